// MyModel_87454124082166
// MI455X (gfx1250) — compile-verified
//
#include <hip/hip_runtime.h>

typedef float v2f __attribute__((ext_vector_type(2)));
typedef float v8f __attribute__((ext_vector_type(8)));

#define T_STEPS 512
#define FEAT    64
#define UNITS   10
#define NROW    16   // batch rows per wave tile
#define HS      20   // LDS h-tile row stride in floats (even -> 8B aligned b64 loads, conflict-free)
#define PF_DIST 4    // prefetch distance (timesteps ahead of compute)

static __device__ inline v8f wmma4(v2f a, v2f b, v8f c) {
  // D = A(16x4 f32) * B(4x16 f32) + C(16x16 f32)
  return __builtin_amdgcn_wmma_f32_16x16x4_f32(false, a, false, b, (short)0, c, false, false);
}

static __device__ inline v8f bc8(float v) {
  v8f r;
#pragma unroll
  for (int i = 0; i < 8; ++i) r[i] = v;
  return r;
}

__global__ __launch_bounds__(32) void gru_fused_wmma(
    const float* __restrict__ x,     // [4096, 512, 64]
    const float* __restrict__ W,     // [64, 30]  (kernel)
    const float* __restrict__ R,     // [10, 30]  (recurrent_kernel)
    const float* __restrict__ bias,  // [2, 30]
    float* __restrict__ out)         // [4096, 10]
{
  __shared__ float hbuf[16 * HS];

  const int lane = threadIdx.x & 31;
  const int u    = lane & 15;   // D-layout column (unit); A-layout row index
  const int half = lane >> 4;
  const int row0 = blockIdx.x * NROW;

  const float uval = (u < UNITS) ? 1.0f : 0.0f;
  const int   cu   = (u < UNITS) ? u : (UNITS - 1);

  // ---- Gather weights into WMMA B-layout registers (once) -------------
  // B-layout per K=4 chunk: lane L, vgpr e holds B[k = 4c + 2*(L>>4) + e][n = L&15]
  v2f Wz[16], Wr[16], Wh[16];
#pragma unroll
  for (int c = 0; c < 16; ++c) {
#pragma unroll
    for (int e = 0; e < 2; ++e) {
      int k = 4 * c + 2 * half + e;
      Wz[c][e] = W[k * 30 + cu]        * uval;
      Wr[c][e] = W[k * 30 + 10 + cu]   * uval;
      Wh[c][e] = W[k * 30 + 20 + cu]   * uval;
    }
  }
  v2f Rz[3], Rr[3], Rh[3];
#pragma unroll
  for (int c = 0; c < 3; ++c) {
#pragma unroll
    for (int e = 0; e < 2; ++e) {
      int   k  = 4 * c + 2 * half + e;
      int   kc = (k < UNITS) ? k : (UNITS - 1);
      float kv = (k < UNITS) ? uval : 0.0f;    // zero padded K rows (10,11)
      Rz[c][e] = R[kc * 30 + cu]        * kv;
      Rr[c][e] = R[kc * 30 + 10 + cu]   * kv;
      Rh[c][e] = R[kc * 30 + 20 + cu]   * kv;
    }
  }
  // biases: z/r gates can fold input+recurrent bias; h keeps them split
  const float bz  = (bias[cu]      + bias[30 + cu]) * uval;
  const float br  = (bias[10 + cu] + bias[40 + cu]) * uval;
  const float bih =  bias[20 + cu] * uval;
  const float brh =  bias[50 + cu] * uval;

  // ---- X loads: directly in A-layout (lane row = u, K pair at 2*half) --
  const float* xrow = x + (size_t)(row0 + u) * T_STEPS * FEAT + 2 * half;
  // prefetch pointer: 32 lanes cover the 16 rows x 256B (= 32 cachelines of 128B)
  const float* xpf  = x + (size_t)(row0 + u) * T_STEPS * FEAT + half * 32;

  v2f xa[16], xb[16];
  v8f hD = bc8(0.0f);     // h in D-layout (rows i+8*half, col u)
  v2f hA[3];              // h in A-layout (K chunks), h0 = 0
#pragma unroll
  for (int c = 0; c < 3; ++c) { hA[c][0] = 0.0f; hA[c][1] = 0.0f; }

  auto loadX = [&](v2f (&buf)[16], int t) {
    const float* p = xrow + (size_t)t * FEAT;
#pragma unroll
    for (int c = 0; c < 16; ++c) buf[c] = *(const v2f*)(p + 4 * c);
  };

  auto step = [&](const v2f (&xc)[16]) {
    v8f Az = bc8(bz), Ar = bc8(br), Axh = bc8(bih), Ahh = bc8(brh);
    // input GEMM: K = 64 -> 16 chunks, 3 gate tiles
#pragma unroll
    for (int c = 0; c < 16; ++c) {
      Az  = wmma4(xc[c], Wz[c], Az);
      Ar  = wmma4(xc[c], Wr[c], Ar);
      Axh = wmma4(xc[c], Wh[c], Axh);
    }
    // recurrent GEMM: K = 12 (10 + zero pad) -> 3 chunks
#pragma unroll
    for (int c = 0; c < 3; ++c) {
      Az  = wmma4(hA[c], Rz[c], Az);
      Ar  = wmma4(hA[c], Rr[c], Ar);
      Ahh = wmma4(hA[c], Rh[c], Ahh);
    }
    // gates: fully elementwise (lane = unit for all three tiles)
#pragma unroll
    for (int i = 0; i < 8; ++i) {
      float z   = 1.0f / (1.0f + __expf(-Az[i]));
      float r   = 1.0f / (1.0f + __expf(-Ar[i]));
      float pre = Axh[i] + r * Ahh[i];
      float th  = 2.0f / (1.0f + __expf(-2.0f * pre)) - 1.0f;   // tanh, inf-safe
      hD[i] = z * hD[i] + (1.0f - z) * th;
    }
    // D-layout -> A-layout transpose via LDS (wave-private tile)
#pragma unroll
    for (int i = 0; i < 8; ++i) hbuf[(i + 8 * half) * HS + u] = hD[i];
    __syncthreads();
#pragma unroll
    for (int c = 0; c < 3; ++c)
      hA[c] = *(const v2f*)&hbuf[u * HS + 4 * c + 2 * half];
    __syncthreads();
  };

  loadX(xa, 0);
  // manual 2x unroll keeps both X buffers in registers (no indirect buffer ptr)
  for (int t = 0; t < T_STEPS; t += 2) {
    // deep prefetch: global_prefetch_b8, no counters, no registers held
    {
      int tp0 = t + PF_DIST;     if (tp0 > T_STEPS - 1) tp0 = T_STEPS - 1;
      int tp1 = t + PF_DIST + 1; if (tp1 > T_STEPS - 1) tp1 = T_STEPS - 1;
      __builtin_prefetch(xpf + (size_t)tp0 * FEAT, 0, 3);
      __builtin_prefetch(xpf + (size_t)tp1 * FEAT, 0, 3);
    }
    loadX(xb, t + 1);
    step(xa);
    if (t + 2 < T_STEPS) loadX(xa, t + 2);
    step(xb);
  }

  if (u < UNITS) {
#pragma unroll
    for (int i = 0; i < 8; ++i)
      out[(size_t)(row0 + i + 8 * half) * UNITS + u] = hD[i];
  }
}

extern "C" void kernel_launch(void* const* d_in, const int* in_sizes, int n_in,
                              void* d_out, int out_size, void* d_ws, size_t ws_size,
                              hipStream_t stream) {
  const float* x  = (const float*)d_in[0];
  const float* W  = (const float*)d_in[1];
  const float* R  = (const float*)d_in[2];
  const float* b  = (const float*)d_in[3];
  float* out      = (float*)d_out;

  const int batch = 4096;
  dim3 grid(batch / NROW);   // 256 independent wave-tiles
  dim3 block(32);            // one wave32 per workgroup
  gru_fused_wmma<<<grid, block, 0, stream>>>(x, W, R, b, out);
}